// YOLOv8Loss_30159260352863
// MI455X (gfx1250) — compile-verified
//
#include <hip/hip_runtime.h>
#include <math.h>

// ---------------- problem constants (match reference setup_inputs) ----------
#define BB 32
#define AA 8400
#define GG 40
#define CC 80
#define RR 16
#define KK 10
#define TPB 256
#define EPSF 1e-7f
#define IMGF 640.0f

#define NB_DEC  ((BB * AA + TPB - 1) / TPB)   // 1050 blocks in k_decode
#define NB_AX   ((AA + TPB - 1) / TPB)        // 33
#define NB_LOSS (NB_AX * BB)                  // 1056 blocks in k_loss

// anchor grid: levels (stride 8, 80x80), (16, 40x40), (32, 20x20)
__device__ __forceinline__ void anchor_of(int a, float& ax, float& ay, float& st) {
  int i, sz; float s;
  if (a < 6400)      { i = a;        s = 8.f;  sz = 80; }
  else if (a < 8000) { i = a - 6400; s = 16.f; sz = 40; }
  else               { i = a - 8000; s = 32.f; sz = 20; }
  ax = ((float)(i % sz) + 0.5f) * s;
  ay = ((float)(i / sz) + 0.5f) * s;
  st = s;
}

__device__ __forceinline__ float pair_iou(float gx1, float gy1, float gx2, float gy2,
                                          float px1, float py1, float px2, float py2) {
  float iw = fminf(gx2, px2) - fmaxf(gx1, px1); iw = fmaxf(iw, 0.f);
  float ih = fminf(gy2, py2) - fmaxf(gy1, py1); ih = fmaxf(ih, 0.f);
  float inter = iw * ih;
  float a1 = (gx2 - gx1) * (gy2 - gy1);
  float a2 = (px2 - px1) * (py2 - py1);
  return inter / (a1 + a2 - inter + EPSF);
}

// deterministic fixed-order block tree reduction (ds traffic, no float atomics)
__device__ __forceinline__ float block_sum(float v, float* red) {
  red[threadIdx.x] = v;
  __syncthreads();
  for (int s = TPB / 2; s > 0; s >>= 1) {
    if (threadIdx.x < s) red[threadIdx.x] += red[threadIdx.x + s];
    __syncthreads();
  }
  float r = red[0];
  __syncthreads();
  return r;
}

// ---------------- kernel 0: zero mask bits ----------------------------------
__global__ void k_init(unsigned long long* __restrict__ mask) {
  int i = blockIdx.x * TPB + threadIdx.x;
  if (i < BB * AA) mask[i] = 0ull;
}

// ---- kernel 1: DFL decode + class max + assignment-independent BCE part ----
__global__ void k_decode(const float* __restrict__ pdist, const float* __restrict__ pcls,
                         float* __restrict__ boxes, float* __restrict__ ssq,
                         float* __restrict__ lse, float* __restrict__ part0) {
  __shared__ float red[TPB];
  int idx = blockIdx.x * TPB + threadIdx.x;
  float softplus_acc = 0.f;
  if (idx < BB * AA) {
    int a = idx % AA;
    const float* dp = pdist + (size_t)idx * 64;
    __builtin_prefetch(pcls + (size_t)idx * CC, 0, 0);   // global_prefetch_b8

    float d[4], l[4];
#pragma unroll
    for (int s_ = 0; s_ < 4; s_++) {
      float x[RR];
      const float4* p4 = (const float4*)(dp + s_ * RR);
#pragma unroll
      for (int q = 0; q < 4; q++) {
        float4 v = p4[q];
        x[q*4+0] = v.x; x[q*4+1] = v.y; x[q*4+2] = v.z; x[q*4+3] = v.w;
      }
      float m = x[0];
#pragma unroll
      for (int r = 1; r < RR; r++) m = fmaxf(m, x[r]);
      float se = 0.f, sw = 0.f;
#pragma unroll
      for (int r = 0; r < RR; r++) { float e = expf(x[r] - m); se += e; sw += e * (float)r; }
      d[s_] = sw / se;
      l[s_] = m + logf(se);
    }
    float ax, ay, st; anchor_of(a, ax, ay, st);
    float bx1 = fminf(fmaxf(ax - d[0] * st, 0.f), IMGF);
    float by1 = fminf(fmaxf(ay - d[1] * st, 0.f), IMGF);
    float bx2 = fminf(fmaxf(ax + d[2] * st, 0.f), IMGF);
    float by2 = fminf(fmaxf(ay + d[3] * st, 0.f), IMGF);
    ((float4*)boxes)[idx] = make_float4(bx1, by1, bx2, by2);
    ((float4*)lse)[idx]   = make_float4(l[0], l[1], l[2], l[3]);

    // class max + Sum_c [max(x,0) + log1p(exp(-|x|))]  (target-independent BCE part)
    const float4* cp4 = (const float4*)(pcls + (size_t)idx * CC);
    float cm = -3.4e38f;
#pragma unroll 4
    for (int q = 0; q < CC / 4; q++) {
      float4 v = cp4[q];
      float xs[4] = { v.x, v.y, v.z, v.w };
#pragma unroll
      for (int e = 0; e < 4; e++) {
        float x = xs[e];
        cm = fmaxf(cm, x);
        softplus_acc += fmaxf(x, 0.f) + log1pf(expf(-fabsf(x)));
      }
    }
    ssq[idx] = sqrtf(1.f / (1.f + expf(-cm)));   // sqrt(sigmoid(max)) = cls_score^ALPHA
  }
  float bs = block_sum(softplus_acc, red);
  if (threadIdx.x == 0) part0[blockIdx.x] = bs;
}

// ---------------- kernel 2: align top-10 per (b,g) -> mask bits -------------
__global__ void k_topk(const float* __restrict__ gtb, const unsigned char* __restrict__ mgt,
                       const float* __restrict__ boxes, const float* __restrict__ ssq,
                       unsigned long long* __restrict__ mask) {
  int bg = blockIdx.x;
  int b = bg / GG, g = bg % GG;
  __shared__ float sGT[4];
  __shared__ float sv[TPB * KK];
  __shared__ int   si[TPB * KK];
  __shared__ float rv[TPB];
  __shared__ int   ri[TPB];

  // CDNA5 async global->LDS staging of the GT box (4 dwords, lanes 0..3)
  if (threadIdx.x < 4) {
    unsigned lds_off = (unsigned)(size_t)(&sGT[threadIdx.x]);
    unsigned voff    = (unsigned)((bg * 4 + threadIdx.x) * 4);
    asm volatile("global_load_async_to_lds_b32 %0, %1, %2"
                 :: "v"(lds_off), "v"(voff), "s"(gtb) : "memory");
  }
  asm volatile("s_wait_asynccnt 0" ::: "memory");
  __syncthreads();
  if (!mgt[bg]) return;   // uniform exit: mask stays zero for this gt

  float gx1 = sGT[0], gy1 = sGT[1], gx2 = sGT[2], gy2 = sGT[3];

  float tv[KK]; int ti[KK];
#pragma unroll
  for (int i = 0; i < KK; i++) { tv[i] = -1.f; ti[i] = 0; }

  const float4* bx = (const float4*)(boxes + (size_t)b * AA * 4);
  const float*  sq = ssq + (size_t)b * AA;
  for (int a = threadIdx.x; a < AA; a += TPB) {
    float4 p = bx[a];
    float iou = pair_iou(gx1, gy1, gx2, gy2, p.x, p.y, p.z, p.w);
    float i2 = iou * iou;
    float al = sq[a] * (i2 * i2 * i2);   // score^0.5 * iou^6
    float v = al; int id = a;
#pragma unroll
    for (int i = 0; i < KK; i++) {       // branch-free sorted insert
      bool gt_ = v > tv[i];
      float nv = gt_ ? v : tv[i]; float ov = gt_ ? tv[i] : v;
      int   ni = gt_ ? id : ti[i]; int  oi = gt_ ? ti[i] : id;
      tv[i] = nv; v = ov; ti[i] = ni; id = oi;
    }
  }
#pragma unroll
  for (int i = 0; i < KK; i++) { sv[threadIdx.x * KK + i] = tv[i]; si[threadIdx.x * KK + i] = ti[i]; }
  __syncthreads();

  for (int sel = 0; sel < KK; sel++) {
    float mv = -3.f; int mj = 0;
    for (int j = threadIdx.x; j < TPB * KK; j += TPB) {
      float v = sv[j]; if (v > mv) { mv = v; mj = j; }
    }
    rv[threadIdx.x] = mv; ri[threadIdx.x] = mj;
    __syncthreads();
    for (int s = TPB / 2; s > 0; s >>= 1) {
      if (threadIdx.x < s && rv[threadIdx.x + s] > rv[threadIdx.x]) {
        rv[threadIdx.x] = rv[threadIdx.x + s]; ri[threadIdx.x] = ri[threadIdx.x + s];
      }
      __syncthreads();
    }
    if (threadIdx.x == 0) {
      int j = ri[0];
      if (rv[0] > 0.f)
        atomicOr(&mask[(size_t)b * AA + (size_t)si[j]], 1ull << g);   // order-independent
      sv[j] = -3.f;
    }
    __syncthreads();
  }
}

// ---------------- kernel 3: per-anchor loss terms ---------------------------
__global__ void k_loss(const float* __restrict__ pdist, const float* __restrict__ pcls,
                       const int* __restrict__ gtl, const float* __restrict__ gtb,
                       const float* __restrict__ boxes, const float* __restrict__ lse,
                       const unsigned long long* __restrict__ mask,
                       float* __restrict__ part1) {
  int b = blockIdx.y;
  int a = blockIdx.x * TPB + threadIdx.x;
  int blin = b * NB_AX + blockIdx.x;
  __shared__ float sGT[GG * 4];
  __shared__ int   sLab[GG];
  __shared__ float red[TPB];

  if (threadIdx.x < GG * 4) {   // async-stage all GT boxes for this batch
    unsigned lds_off = (unsigned)(size_t)(&sGT[threadIdx.x]);
    unsigned voff    = (unsigned)((b * GG * 4 + threadIdx.x) * 4);
    asm volatile("global_load_async_to_lds_b32 %0, %1, %2"
                 :: "v"(lds_off), "v"(voff), "s"(gtb) : "memory");
  }
  if (threadIdx.x < GG) sLab[threadIdx.x] = gtl[b * GG + threadIdx.x];
  asm volatile("s_wait_asynccnt 0" ::: "memory");
  __syncthreads();

  float c_clsneg = 0.f, c_score = 0.f, c_nfg = 0.f, c_box = 0.f, c_dfl = 0.f;
  if (a < AA) {
    int idx = b * AA + a;
    unsigned long long mm = mask[idx];
    if (mm != 0ull) {                              // foreground anchor
      float4 pb = ((const float4*)boxes)[idx];
      float best = 0.f; int bg_ = 0;
      for (int g = 0; g < GG; g++) {
        if (mm & (1ull << g)) {
          float iou = pair_iou(sGT[g*4+0], sGT[g*4+1], sGT[g*4+2], sGT[g*4+3],
                               pb.x, pb.y, pb.z, pb.w);
          if (iou > best) { best = iou; bg_ = g; }
        }
      }
      float iou_s = best;
      int tlab = sLab[bg_];
      tlab = tlab < 0 ? 0 : (tlab > CC - 1 ? CC - 1 : tlab);

      // target-dependent BCE part: -x[tlab] * t  (single scalar load)
      c_clsneg = -pcls[(size_t)idx * CC + tlab] * iou_s;
      c_score  = iou_s;
      c_nfg    = 1.f;

      float tx1 = sGT[bg_*4+0], ty1 = sGT[bg_*4+1], tx2 = sGT[bg_*4+2], ty2 = sGT[bg_*4+3];
      // CIoU(pred=b1, target=b2)
      float iw = fminf(pb.z, tx2) - fmaxf(pb.x, tx1); iw = fmaxf(iw, 0.f);
      float ih = fminf(pb.w, ty2) - fmaxf(pb.y, ty1); ih = fmaxf(ih, 0.f);
      float inter = iw * ih;
      float w1 = fmaxf(pb.z - pb.x, EPSF), h1 = fmaxf(pb.w - pb.y, EPSF);
      float w2 = fmaxf(tx2 - tx1, EPSF),   h2 = fmaxf(ty2 - ty1, EPSF);
      float uni = w1 * h1 + w2 * h2 - inter + EPSF;
      float iou = inter / uni;
      float cw = fmaxf(pb.z, tx2) - fminf(pb.x, tx1);
      float ch = fmaxf(pb.w, ty2) - fminf(pb.y, ty1);
      float c2 = cw * cw + ch * ch + EPSF;
      float dx = (pb.x + pb.z) * 0.5f - (tx1 + tx2) * 0.5f;
      float dy = (pb.y + pb.w) * 0.5f - (ty1 + ty2) * 0.5f;
      float rho2 = dx * dx + dy * dy;
      const float PI = 3.14159265358979323846f;
      float dv = atanf(w2 / h2) - atanf(w1 / h1);
      float vv = (4.f / (PI * PI)) * dv * dv;
      float alpha = vv / (1.f - iou + vv + EPSF);
      float ci = iou - (rho2 / c2 + vv * alpha);
      ci = fminf(fmaxf(ci, -1.f), 1.f);
      c_box = 1.f - ci;

      // DFL
      float ax, ay, st; anchor_of(a, ax, ay, st);
      float4 l4 = ((const float4*)lse)[idx];
      float ls[4] = { l4.x, l4.y, l4.z, l4.w };
      float tg[4] = { (ax - tx1) / st, (ay - ty1) / st, (tx2 - ax) / st, (ty2 - ay) / st };
      const float* dp = pdist + (size_t)idx * 64;
      float dfl = 0.f;
#pragma unroll
      for (int s_ = 0; s_ < 4; s_++) {
        float t = fminf(fmaxf(tg[s_], 0.f), (float)RR - 1.01f);
        int tl = (int)floorf(t);
        tl = tl < 0 ? 0 : (tl > RR - 1 ? RR - 1 : tl);
        int tr = tl + 1; tr = tr > RR - 1 ? RR - 1 : tr;
        float wr = fminf(fmaxf(t - (float)tl, 0.f), 1.f);
        float wl = 1.f - wr;
        float xl = dp[s_ * RR + tl];
        float xr = dp[s_ * RR + tr];
        dfl += (ls[s_] - xl) * wl + (ls[s_] - xr) * wr;   // -(logp)
      }
      c_dfl = dfl;
    }
  }
  // deterministic per-block reduction -> partials
  float s0 = block_sum(c_clsneg, red);
  float s1 = block_sum(c_score, red);
  float s2 = block_sum(c_nfg, red);
  float s3 = block_sum(c_box, red);
  float s4 = block_sum(c_dfl, red);
  if (threadIdx.x == 0) {
    part1[0 * NB_LOSS + blin] = s0;
    part1[1 * NB_LOSS + blin] = s1;
    part1[2 * NB_LOSS + blin] = s2;
    part1[3 * NB_LOSS + blin] = s3;
    part1[4 * NB_LOSS + blin] = s4;
  }
}

// ---------------- kernel 4: deterministic final reduction + combine ---------
__global__ void k_final(const float* __restrict__ part0, const float* __restrict__ part1,
                        float* __restrict__ out) {
  __shared__ float red[TPB];
  float v = 0.f;
  for (int i = threadIdx.x; i < NB_DEC; i += TPB) v += part0[i];
  float softplus = block_sum(v, red);
  float terms[5];
  for (int t = 0; t < 5; t++) {
    float s = 0.f;
    for (int i = threadIdx.x; i < NB_LOSS; i += TPB) s += part1[t * NB_LOSS + i];
    terms[t] = block_sum(s, red);
  }
  if (threadIdx.x == 0) {
    float cls = softplus + terms[0];      // Σ softplus - Σ x*t
    float sc  = fmaxf(terms[1], 1.f);     // score_sum
    float nf  = fmaxf(terms[2], 1.f);     // num_fg
    out[0] = 7.5f * (terms[3] / nf) + 0.5f * (cls / sc) + 1.5f * ((terms[4] / nf) * 0.25f);
  }
}

extern "C" void kernel_launch(void* const* d_in, const int* in_sizes, int n_in,
                              void* d_out, int out_size, void* d_ws, size_t ws_size,
                              hipStream_t stream) {
  const float* pdist = (const float*)d_in[0];               // (B,A,64)
  const float* pcls  = (const float*)d_in[1];               // (B,A,80)
  const int*   gtl   = (const int*)d_in[2];                 // (B,G)
  const float* gtb   = (const float*)d_in[3];               // (B,G,4)
  const unsigned char* mgt = (const unsigned char*)d_in[4]; // (B,G) bool

  float* ws    = (float*)d_ws;
  float* boxes = ws;                                        // B*A*4
  float* ssq   = boxes + (size_t)BB * AA * 4;               // B*A
  float* lseb  = ssq + (size_t)BB * AA;                     // B*A*4
  unsigned long long* mask = (unsigned long long*)(lseb + (size_t)BB * AA * 4); // B*A
  float* part0 = (float*)(mask + (size_t)BB * AA);          // NB_DEC
  float* part1 = part0 + NB_DEC;                            // 5*NB_LOSS

  int n = BB * AA;
  int blocks = (n + TPB - 1) / TPB;
  k_init  <<<blocks, TPB, 0, stream>>>(mask);
  k_decode<<<blocks, TPB, 0, stream>>>(pdist, pcls, boxes, ssq, lseb, part0);
  k_topk  <<<BB * GG, TPB, 0, stream>>>(gtb, mgt, boxes, ssq, mask);
  dim3 g3(NB_AX, BB);
  k_loss  <<<g3, TPB, 0, stream>>>(pdist, pcls, gtl, gtb, boxes, lseb, mask, part1);
  k_final <<<1, TPB, 0, stream>>>(part0, part1, (float*)d_out);
}